// GridExp_60799557042564
// MI455X (gfx1250) — compile-verified
//
#include <hip/hip_runtime.h>
#include <stdint.h>

// Velocity-field exponentiation (scaling & squaring), 8 steps.
// v <- v + trilerp(v, id + v)  with circular boundary; out = id + v.
// Shape (2,160,160,160,3) f32.  Bandwidth-bound gather kernel; the whole
// 98 MB field fits in MI455X's 192 MB L2, so gather amplification is absorbed.

#define DIMV    160
#define NBATCH  2
#define NVOX    (NBATCH * DIMV * DIMV * DIMV)   // 8,192,000 (divisible by 256)
#define THREADS 256

struct F3 { float x, y, z; };

__device__ __forceinline__ int wrap160(int a) {
    int r = a % DIMV;
    return r < 0 ? r + DIMV : r;
}

__global__ __launch_bounds__(THREADS) void vexp_step(
    const float* __restrict__ src, float* __restrict__ dst,
    float scale, int add_grid)
{
    const int i = blockIdx.x * THREADS + threadIdx.x;   // flat voxel id

    // ---- CDNA5 async global->LDS staging of this thread's own vector ----
    // The 12-byte self-read is copied by the async engine while the wave
    // does the integer index decomposition below; one s_wait_asynccnt 0
    // before consuming it from LDS.
    __shared__ float smem[THREADS * 3];
    uint32_t lds_off = (uint32_t)(uintptr_t)(&smem[threadIdx.x * 3]);
    uint32_t goff    = (uint32_t)i * 12u;               // byte offset into src
    asm volatile(
        "global_load_async_to_lds_b32 %0, %1, %2\n\t"
        "global_load_async_to_lds_b32 %0, %1, %2 offset:4\n\t"
        "global_load_async_to_lds_b32 %0, %1, %2 offset:8"
        :
        : "v"(lds_off), "v"(goff), "s"(src)
        : "memory");

    // index math overlaps the async copy
    int z = i % DIMV;
    int t = i / DIMV;
    int y = t % DIMV;
    t     = t / DIMV;
    int x = t % DIMV;
    int b = t / DIMV;

    asm volatile("s_wait_asynccnt 0" ::: "memory");
    float vx = scale * smem[threadIdx.x * 3 + 0];
    float vy = scale * smem[threadIdx.x * 3 + 1];
    float vz = scale * smem[threadIdx.x * 3 + 2];

    // phi = id + v
    float px = (float)x + vx;
    float py = (float)y + vy;
    float pz = (float)z + vz;

    float fx = floorf(px), fy = floorf(py), fz = floorf(pz);
    float tx = px - fx,    ty = py - fy,    tz = pz - fz;

    int x0 = wrap160((int)fx),     y0 = wrap160((int)fy),     z0 = wrap160((int)fz);
    int x1 = wrap160((int)fx + 1), y1 = wrap160((int)fy + 1), z1 = wrap160((int)fz + 1);

    float wx0 = 1.0f - tx, wx1 = tx;
    float wy0 = 1.0f - ty, wy1 = ty;
    float wz0 = 1.0f - tz, wz1 = tz;

    // voxel index = ((b*160 + x)*160 + y)*160 + z
    const F3* __restrict__ s3 = (const F3*)src;
    int bx0 = (b * DIMV + x0) * DIMV;
    int bx1 = (b * DIMV + x1) * DIMV;
    int r00 = (bx0 + y0) * DIMV;
    int r01 = (bx0 + y1) * DIMV;
    int r10 = (bx1 + y0) * DIMV;
    int r11 = (bx1 + y1) * DIMV;

    float ax = 0.0f, ay = 0.0f, az = 0.0f;
#define CORNER(rb, zz, w) do { F3 g = s3[(rb) + (zz)]; float wc = (w); \
        ax = fmaf(wc, g.x, ax); ay = fmaf(wc, g.y, ay); az = fmaf(wc, g.z, az); } while (0)
    CORNER(r00, z0, wx0 * wy0 * wz0);
    CORNER(r00, z1, wx0 * wy0 * wz1);
    CORNER(r01, z0, wx0 * wy1 * wz0);
    CORNER(r01, z1, wx0 * wy1 * wz1);
    CORNER(r10, z0, wx1 * wy0 * wz0);
    CORNER(r10, z1, wx1 * wy0 * wz1);
    CORNER(r11, z0, wx1 * wy1 * wz0);
    CORNER(r11, z1, wx1 * wy1 * wz1);
#undef CORNER

    // v_new = v + sample(v, phi)  (gathered values carry the same scale)
    float ox = fmaf(scale, ax, vx);
    float oy = fmaf(scale, ay, vy);
    float oz = fmaf(scale, az, vz);

    if (add_grid) {            // final step: out = id + v
        ox += (float)x;
        oy += (float)y;
        oz += (float)z;
    }

    F3* __restrict__ d3 = (F3*)dst;
    F3 o; o.x = ox; o.y = oy; o.z = oz;
    d3[i] = o;
}

extern "C" void kernel_launch(void* const* d_in, const int* in_sizes, int n_in,
                              void* d_out, int out_size, void* d_ws, size_t ws_size,
                              hipStream_t stream) {
    (void)in_sizes; (void)n_in; (void)out_size; (void)ws_size;
    const float* vel = (const float*)d_in[0];
    float*       out = (float*)d_out;
    float*       ws  = (float*)d_ws;     // one 98.3 MB ping buffer

    dim3 grid(NVOX / THREADS);

    // step 0: v0 = velocity / 2^8 folded into the gather via `scale`
    vexp_step<<<grid, THREADS, 0, stream>>>(vel, ws, 1.0f / 256.0f, 0);

    // steps 1..7: ping-pong ws <-> out; step 7 lands in out with +grid
    for (int k = 1; k < 8; ++k) {
        const float* s = (k & 1) ? ws  : out;
        float*       d = (k & 1) ? out : ws;
        vexp_step<<<grid, THREADS, 0, stream>>>(s, d, 1.0f, (k == 7) ? 1 : 0);
    }
}